// RotationZ_38079180046915
// MI455X (gfx1250) — compile-verified
//
#include <hip/hip_runtime.h>

// RotationZ: out[b,0] = px*c + py*s ; out[b,1] = -px*s + py*c ; out[b,2] = pz
// alpha_rot[b] = 2*pi*alpha[b]  (appended after the main tensor in d_out)
//
// HBM-bound: ~369 MB moved, ~0.12 GFLOP -> ~15.8 us floor @ 23.3 TB/s.
//  - x/y: direct b128 non-temporal load->rotate->store (touched once, keep
//    out of the 192 MB L2).
//  - z:   pure copy -> routed through the CDNA5 async LDS DMA path
//         (global_load_async_to_lds_b128 / global_store_async_from_lds_b128,
//         ASYNCcnt), overlapped with the x/y math, zero VGPR traffic.
//  - alpha uniform per block -> scalar load, one sincos per thread.

typedef float v4f __attribute__((ext_vector_type(4)));
typedef int   v4i __attribute__((ext_vector_type(4)));

#define TWO_PI_F 6.283185307179586476925286766559f

#if defined(__AMDGCN__) &&                                            \
    __has_builtin(__builtin_amdgcn_global_load_async_to_lds_b128) &&  \
    __has_builtin(__builtin_amdgcn_global_store_async_from_lds_b128)
#define USE_ASYNC_LDS 1
#else
#define USE_ASYNC_LDS 0
#endif

#if USE_ASYNC_LDS
// Builtin signature (from hipcc diagnostic): int4 pointers, AS1 (global,
// printed as __device__) and AS3 (LDS), plus two immediate ints.
typedef __attribute__((address_space(1))) v4i g_v4i;   // global int4
typedef __attribute__((address_space(3))) v4i l_v4i;   // LDS int4

__device__ __forceinline__ void wait_asynccnt0() {
#if __has_builtin(__builtin_amdgcn_s_wait_asynccnt)
    __builtin_amdgcn_s_wait_asynccnt(0);
#else
    asm volatile("s_wait_asynccnt 0x0" ::: "memory");
#endif
}
#endif

__global__ __launch_bounds__(256) void rotz_stream_kernel(
    const float* __restrict__ x,
    const float* __restrict__ alpha,
    float* __restrict__ out,
    float* __restrict__ alpha_rot,
    int plane,    // 300*25 = 7500 floats per channel plane
    int plane4)   // plane/4 = 1875 float4 per plane
{
    const int b   = blockIdx.x;
    const int tid = threadIdx.x;

    // Uniform per block -> s_load + scalar multiply.
    const float ar = alpha[b] * TWO_PI_F;
    if (tid == 0) {
        alpha_rot[b] = ar;
    }

    float s, c;
    __sincosf(ar, &s, &c);   // v_sin_f32 / v_cos_f32, once per thread

    const size_t base = (size_t)b * 3u * (size_t)plane;
    const v4f* __restrict__ px = (const v4f*)(x + base);
    const v4f* __restrict__ py = px + plane4;
    const v4f* __restrict__ pz = py + plane4;
    v4f* __restrict__ ox = (v4f*)(out + base);
    v4f* __restrict__ oy = ox + plane4;
    v4f* __restrict__ oz = oy + plane4;

#if USE_ASYNC_LDS
    // Kick off the whole z plane (plane4*16 = 30000 B) global->LDS via the
    // async DMA path. Runs concurrently with the x/y rotation stream below.
    __shared__ __align__(16) unsigned char zbuf[30016];
    for (int i = tid; i < plane4; i += 256) {
        __builtin_amdgcn_global_load_async_to_lds_b128(
            (g_v4i*)(pz + i), (l_v4i*)(zbuf + (size_t)i * 16),
            /*offset=*/0, /*cpol=*/0);
    }
#endif

    // x/y rotation: scalar trip count (no per-iteration EXEC churn),
    // 7 full 256-lane sweeps + one predicated tail for plane4 = 1875.
    const int nfull = plane4 >> 8;
    int i = tid;
    #pragma unroll 2
    for (int k = 0; k < nfull; ++k, i += 256) {
        v4f vx = __builtin_nontemporal_load(px + i);
        v4f vy = __builtin_nontemporal_load(py + i);
        v4f rx = vx * c + vy * s;
        v4f ry = vy * c - vx * s;
        __builtin_nontemporal_store(rx, ox + i);
        __builtin_nontemporal_store(ry, oy + i);
    }
    if (i < plane4) {
        v4f vx = __builtin_nontemporal_load(px + i);
        v4f vy = __builtin_nontemporal_load(py + i);
        v4f rx = vx * c + vy * s;
        v4f ry = vy * c - vx * s;
        __builtin_nontemporal_store(rx, ox + i);
        __builtin_nontemporal_store(ry, oy + i);
    }

#if USE_ASYNC_LDS
    // Drain this wave's async loads; each thread stores only slots its own
    // wave loaded, so no block barrier is required. S_ENDPGM's implicit
    // wait-idle covers the stores.
    wait_asynccnt0();
    for (int j = tid; j < plane4; j += 256) {
        __builtin_amdgcn_global_store_async_from_lds_b128(
            (g_v4i*)(oz + j), (l_v4i*)(zbuf + (size_t)j * 16),
            /*offset=*/0, /*cpol=*/0);
    }
#else
    for (int j = tid; j < plane4; j += 256) {
        v4f vz = __builtin_nontemporal_load(pz + j);
        __builtin_nontemporal_store(vz, oz + j);
    }
#endif
}

extern "C" void kernel_launch(void* const* d_in, const int* in_sizes, int n_in,
                              void* d_out, int out_size, void* d_ws, size_t ws_size,
                              hipStream_t stream) {
    const float* x     = (const float*)d_in[0];   // (2048, 3, 300, 25) f32
    const float* alpha = (const float*)d_in[1];   // (2048,) f32
    float* out = (float*)d_out;

    const int nb     = in_sizes[1];               // 2048
    const int plane  = in_sizes[0] / (3 * nb);    // 7500
    const int plane4 = plane / 4;                 // 1875 (7500 % 4 == 0)

    float* alpha_rot = out + (size_t)in_sizes[0]; // tuple tail: (2048,) f32

    rotz_stream_kernel<<<dim3(nb), dim3(256), 0, stream>>>(
        x, alpha, out, alpha_rot, plane, plane4);
}